// RGGCN_8564164788984
// MI455X (gfx1250) — compile-verified
//
#include <hip/hip_runtime.h>
#include <hip/hip_bf16.h>
#include <math.h>

typedef __attribute__((ext_vector_type(2))) float v2f;
typedef __attribute__((ext_vector_type(8))) float v8f;

#define N_GRAPHS 64
#define OUT_C    64
#define N_CLS    2

__device__ __forceinline__ float sigmoidf_fast(float x) {
    return 1.0f / (1.0f + __expf(-x));
}

// ---------------------------------------------------------------------------
// OUT[N x Cout] = IN[N x Cin] @ W[Cout x Cin]^T + bias
// One wave per 16x16 output tile; K-dim consumed 4 at a time via
// V_WMMA_F32_16X16X4_F32 (fp32 matrix pipe, wave32).
// A layout: lanes 0-15 rows M=0..15 holding K={k,k+1}; lanes 16-31 K={k+2,k+3}.
// B layout: VGPR0 lanes0-15 = row K=k (N=0..15), lanes16-31 = row K=k+2; VGPR1 K=k+1/k+3.
// C/D layout: VGPR r, lanes 0-15 -> D[r][n], lanes 16-31 -> D[r+8][n].
// ---------------------------------------------------------------------------
__global__ __launch_bounds__(32) void gemm_wmma_f32(
    const float* __restrict__ A, const float* __restrict__ W,
    const float* __restrict__ bias, float* __restrict__ out,
    int N, int Cin, int Cout)
{
    const int lane = threadIdx.x & 31;
    const int hi   = lane >> 4;    // 0: lanes 0-15, 1: lanes 16-31
    const int lo   = lane & 15;
    const int rowBase = blockIdx.x * 16;
    const int colBase = blockIdx.y * 16;

    int arow_i = rowBase + lo;
    if (arow_i >= N) arow_i = N - 1;              // clamp (EXEC must stay all-1 for WMMA)
    const float* arow = A + (size_t)arow_i * Cin + 2 * hi;
    const float* wrow = W + (size_t)(colBase + lo) * Cin + 2 * hi;

    const float bb = bias[colBase + lo];
    v8f c = {bb, bb, bb, bb, bb, bb, bb, bb};

    for (int k = 0; k < Cin; k += 4) {
        v2f a, b;
        a.x = arow[k];     a.y = arow[k + 1];
        b.x = wrow[k];     b.y = wrow[k + 1];
        c = __builtin_amdgcn_wmma_f32_16x16x4_f32(
                /*neg_a=*/false, a, /*neg_b=*/false, b,
                /*c_mod=*/(short)0, c, /*reuse_a=*/false, /*reuse_b=*/false);
    }

    #pragma unroll
    for (int r = 0; r < 8; ++r) {
        const int orow = rowBase + 8 * hi + r;
        if (orow < N)
            out[(size_t)orow * Cout + colBase + lo] = c[r];
    }
}

// ---------------------------------------------------------------------------
// Edge phase: eta = sigmoid(K[dst] + Q[src]); H[dst] += eta * V[src]
// One lane handles 4 channels of one edge (float4 gathers, f32 global atomics).
// shift = log2(C/4): 5 for C=128, 4 for C=64.
// ---------------------------------------------------------------------------
__global__ void edge_gate(const float* __restrict__ K, const float* __restrict__ Q,
                          const float* __restrict__ V, float* __restrict__ H,
                          const int* __restrict__ src, const int* __restrict__ dst,
                          int E, int C, int shift)
{
    const int gid  = blockIdx.x * blockDim.x + threadIdx.x;
    const int e    = gid >> shift;
    const int l    = gid & ((1 << shift) - 1);
    if (e >= E) return;

    const int s = src[e];
    const int d = dst[e];
    const size_t off = (size_t)l << 2;

    const float4 kk = *(const float4*)(K + (size_t)d * C + off);
    const float4 qq = *(const float4*)(Q + (size_t)s * C + off);
    const float4 vv = *(const float4*)(V + (size_t)s * C + off);

    float* hp = H + (size_t)d * C + off;
    atomicAdd(hp + 0, vv.x * sigmoidf_fast(kk.x + qq.x));
    atomicAdd(hp + 1, vv.y * sigmoidf_fast(kk.y + qq.y));
    atomicAdd(hp + 2, vv.z * sigmoidf_fast(kk.z + qq.z));
    atomicAdd(hp + 3, vv.w * sigmoidf_fast(kk.w + qq.w));
}

// In-place ReLU (float4)
__global__ void relu_inplace(float* __restrict__ h, int n4)
{
    const int gid = blockIdx.x * blockDim.x + threadIdx.x;
    if (gid >= n4) return;
    float4 v = ((float4*)h)[gid];
    v.x = fmaxf(v.x, 0.f); v.y = fmaxf(v.y, 0.f);
    v.z = fmaxf(v.z, 0.f); v.w = fmaxf(v.w, 0.f);
    ((float4*)h)[gid] = v;
}

__global__ void zero_f32(float* __restrict__ p, int n)
{
    const int gid = blockIdx.x * blockDim.x + threadIdx.x;
    if (gid < n) p[gid] = 0.0f;
}

// Mean-pool (sums + counts) over sorted batch ids; one wave per node (C=64 -> 2 ch/lane)
__global__ void pool_sum(const float* __restrict__ h, const int* __restrict__ batch,
                         float* __restrict__ pool, float* __restrict__ cnt, int N)
{
    const int gid  = blockIdx.x * blockDim.x + threadIdx.x;
    const int node = gid >> 5;
    const int lane = gid & 31;
    if (node >= N) return;
    const int g = batch[node];
    const float2 v = *(const float2*)(h + (size_t)node * OUT_C + lane * 2);
    atomicAdd(pool + (size_t)g * OUT_C + lane * 2 + 0, v.x);
    atomicAdd(pool + (size_t)g * OUT_C + lane * 2 + 1, v.y);
    if (lane == 0) atomicAdd(cnt + g, 1.0f);
}

// out[g, cls] = bfc[cls] + sum_c (pool[g,c]/max(cnt[g],1)) * wfc[cls,c]
__global__ void fc_head(const float* __restrict__ pool, const float* __restrict__ cnt,
                        const float* __restrict__ wfc, const float* __restrict__ bfc,
                        float* __restrict__ out)
{
    const int t = blockIdx.x * blockDim.x + threadIdx.x;
    if (t >= N_GRAPHS * N_CLS) return;
    const int g   = t / N_CLS;
    const int cls = t % N_CLS;
    const float inv = 1.0f / fmaxf(cnt[g], 1.0f);
    float acc = bfc[cls];
    #pragma unroll 8
    for (int c = 0; c < OUT_C; ++c)
        acc += pool[g * OUT_C + c] * inv * wfc[cls * OUT_C + c];
    out[t] = acc;
}

extern "C" void kernel_launch(void* const* d_in, const int* in_sizes, int n_in,
                              void* d_out, int out_size, void* d_ws, size_t ws_size,
                              hipStream_t stream)
{
    const float* x     = (const float*)d_in[0];
    const int*   ei    = (const int*)d_in[1];
    const int*   batch = (const int*)d_in[2];
    const float* w1k = (const float*)d_in[3];  const float* b1k = (const float*)d_in[4];
    const float* w1q = (const float*)d_in[5];  const float* b1q = (const float*)d_in[6];
    const float* w1v = (const float*)d_in[7];  const float* b1v = (const float*)d_in[8];
    const float* w1s = (const float*)d_in[9];  const float* b1s = (const float*)d_in[10];
    const float* w2k = (const float*)d_in[11]; const float* b2k = (const float*)d_in[12];
    const float* w2q = (const float*)d_in[13]; const float* b2q = (const float*)d_in[14];
    const float* w2v = (const float*)d_in[15]; const float* b2v = (const float*)d_in[16];
    const float* w2s = (const float*)d_in[17]; const float* b2s = (const float*)d_in[18];
    const float* wfc = (const float*)d_in[19]; const float* bfc = (const float*)d_in[20];
    float* out = (float*)d_out;

    const int IN_C  = 128;
    const int HID_C = 128;
    const int N = in_sizes[0] / IN_C;        // 50000
    const int E = in_sizes[1] / 2;           // 800000
    const int* src = ei;
    const int* dst = ei + E;

    // Workspace layout (floats)
    float* ws  = (float*)d_ws;
    float* K1  = ws;                                  // N*128
    float* Q1  = K1 + (size_t)N * HID_C;
    float* V1  = Q1 + (size_t)N * HID_C;
    float* H1  = V1 + (size_t)N * HID_C;              // skip-init, then += messages
    float* K2  = H1 + (size_t)N * HID_C;              // N*64
    float* Q2  = K2 + (size_t)N * OUT_C;
    float* V2  = Q2 + (size_t)N * OUT_C;
    float* H2  = V2 + (size_t)N * OUT_C;
    float* POOL = H2 + (size_t)N * OUT_C;             // 64*64
    float* CNT  = POOL + N_GRAPHS * OUT_C;            // 64

    const int rowTiles = (N + 15) / 16;

    // ---- Layer 1: four GEMMs (128 -> 128); S writes directly into H1 ----
    dim3 g1(rowTiles, HID_C / 16);
    gemm_wmma_f32<<<g1, 32, 0, stream>>>(x, w1k, b1k, K1, N, IN_C, HID_C);
    gemm_wmma_f32<<<g1, 32, 0, stream>>>(x, w1q, b1q, Q1, N, IN_C, HID_C);
    gemm_wmma_f32<<<g1, 32, 0, stream>>>(x, w1v, b1v, V1, N, IN_C, HID_C);
    gemm_wmma_f32<<<g1, 32, 0, stream>>>(x, w1s, b1s, H1, N, IN_C, HID_C);

    // ---- Layer 1 edge phase: 32 lanes per edge (C=128) ----
    {
        const long long total = (long long)E << 5;
        const int blocks = (int)((total + 255) / 256);
        edge_gate<<<blocks, 256, 0, stream>>>(K1, Q1, V1, H1, src, dst, E, HID_C, 5);
    }
    relu_inplace<<<((N * HID_C / 4) + 255) / 256, 256, 0, stream>>>(H1, N * HID_C / 4);

    // ---- Layer 2: four GEMMs (128 -> 64); S writes into H2 ----
    dim3 g2(rowTiles, OUT_C / 16);
    gemm_wmma_f32<<<g2, 32, 0, stream>>>(H1, w2k, b2k, K2, N, HID_C, OUT_C);
    gemm_wmma_f32<<<g2, 32, 0, stream>>>(H1, w2q, b2q, Q2, N, HID_C, OUT_C);
    gemm_wmma_f32<<<g2, 32, 0, stream>>>(H1, w2v, b2v, V2, N, HID_C, OUT_C);
    gemm_wmma_f32<<<g2, 32, 0, stream>>>(H1, w2s, b2s, H2, N, HID_C, OUT_C);

    // ---- Layer 2 edge phase: 16 lanes per edge (C=64) ----
    {
        const long long total = (long long)E << 4;
        const int blocks = (int)((total + 255) / 256);
        edge_gate<<<blocks, 256, 0, stream>>>(K2, Q2, V2, H2, src, dst, E, OUT_C, 4);
    }
    relu_inplace<<<((N * OUT_C / 4) + 255) / 256, 256, 0, stream>>>(H2, N * OUT_C / 4);

    // ---- Global mean pool + FC head ----
    const int poolElems = N_GRAPHS * OUT_C + N_GRAPHS;
    zero_f32<<<(poolElems + 255) / 256, 256, 0, stream>>>(POOL, poolElems);
    {
        const long long total = (long long)N << 5;
        const int blocks = (int)((total + 255) / 256);
        pool_sum<<<blocks, 256, 0, stream>>>(H2, batch, POOL, CNT, N);
    }
    fc_head<<<1, N_GRAPHS * N_CLS, 0, stream>>>(POOL, CNT, wfc, bfc, out);
}